// graph_function_66726611911291
// MI455X (gfx1250) — compile-verified
//
#include <hip/hip_runtime.h>
#include <hip/hip_bf16.h>

typedef __attribute__((ext_vector_type(2))) float v2f;
typedef __attribute__((ext_vector_type(8))) float v8f;

#define N_NODES 100000
#define N_EDGES 640000
#define DFEAT   128
#define BN_EPS  1e-5f

// ---------------------------------------------------------------------------
// Kernel 0: zero workspace (agg + cnt + bn sums)
// ---------------------------------------------------------------------------
__global__ void zero_ws_kernel(float* __restrict__ ws, long n) {
    long i = (long)blockIdx.x * blockDim.x + threadIdx.x;
    long stride = (long)gridDim.x * blockDim.x;
    for (; i < n; i += stride) ws[i] = 0.0f;
}

// ---------------------------------------------------------------------------
// Kernel 1: edge scatter. One wave (32 lanes) per edge; each lane handles a
// float4 slice of the 128-d feature, applying ReLU to the gathered source row
// and atomically accumulating into agg[dst]. Lane 0 bumps the degree count.
// ---------------------------------------------------------------------------
__global__ void __launch_bounds__(256) scatter_edges_kernel(
    const float* __restrict__ x, const int* __restrict__ ei,
    float* __restrict__ agg, float* __restrict__ cnt)
{
    long gtid = (long)blockIdx.x * blockDim.x + threadIdx.x;
    int e    = (int)(gtid >> 5);
    int lane = (int)(gtid & 31);
    if (e >= N_EDGES) return;
    int s = ei[e];              // src row of edge_index
    int d = ei[N_EDGES + e];    // dst row of edge_index
    const float4 v = *reinterpret_cast<const float4*>(x + (long)s * DFEAT + lane * 4);
    float* base = agg + (long)d * DFEAT + lane * 4;
    atomicAdd(base + 0, fmaxf(v.x, 0.0f));
    atomicAdd(base + 1, fmaxf(v.y, 0.0f));
    atomicAdd(base + 2, fmaxf(v.z, 0.0f));
    atomicAdd(base + 3, fmaxf(v.w, 0.0f));
    if (lane == 0) atomicAdd(cnt + d, 1.0f);
}

// ---------------------------------------------------------------------------
// Kernel 2: SAGE GEMM with fp32 WMMA.
// Block = 256 threads = 8 waves, owns a 16-row strip of nodes.
// LDS stages: sA = mean-scaled aggregate (16x128), sH = relu(x) (16x128).
// Wave w computes output column tile j=w as a 16x16 f32 accumulator:
//   C += A_mean(16x4) x W_l^T(4x16)  and  C += H(16x4) x W_r^T(4x16)
// over 32 K-steps -> 64 v_wmma_f32_16x16x4_f32 per wave.
// Epilogue: +bias, store to d_out, atomic per-feature sum / sumsq for BN.
// ---------------------------------------------------------------------------
__global__ void __launch_bounds__(256) sage_gemm_kernel(
    const float* __restrict__ x,   const float* __restrict__ Wl,
    const float* __restrict__ bl,  const float* __restrict__ Wr,
    const float* __restrict__ agg, const float* __restrict__ cnt,
    float* __restrict__ out, float* __restrict__ bnsum, float* __restrict__ bnsq)
{
    __shared__ float sA[16 * DFEAT];   // mean aggregate, pre-scaled
    __shared__ float sH[16 * DFEAT];   // relu(x)
    __shared__ float sInv[16];

    const int  tid  = threadIdx.x;
    const long row0 = (long)blockIdx.x * 16;

    if (tid < 16) {
        float c = cnt[row0 + tid];
        sInv[tid] = 1.0f / fmaxf(c, 1.0f);
    }
    __syncthreads();

    // Stage A matrices: 512 float4 per matrix, 2 per thread.
    #pragma unroll
    for (int it = 0; it < 2; ++it) {
        int f4   = tid + it * 256;       // 0..511
        int row  = f4 >> 5;              // 32 float4 per 128-wide row
        int col4 = (f4 & 31) * 4;
        long g   = (row0 + row) * DFEAT + col4;
        float4 a = *reinterpret_cast<const float4*>(agg + g);
        float4 h = *reinterpret_cast<const float4*>(x + g);
        float sc = sInv[row];
        float4 as = make_float4(a.x * sc, a.y * sc, a.z * sc, a.w * sc);
        float4 hr = make_float4(fmaxf(h.x, 0.0f), fmaxf(h.y, 0.0f),
                                fmaxf(h.z, 0.0f), fmaxf(h.w, 0.0f));
        *reinterpret_cast<float4*>(&sA[row * DFEAT + col4]) = as;
        *reinterpret_cast<float4*>(&sH[row * DFEAT + col4]) = hr;
    }
    __syncthreads();

    const int wave = tid >> 5;       // column tile j = wave (0..7)
    const int lane = tid & 31;
    const int half = lane >> 4;      // K half-selector for 32-bit A/B frags
    const int mn   = lane & 15;      // M for A-frag, N for B/C/D-frag
    const int ng   = wave * 16 + mn; // global output column

    const float* wl_row = Wl + (long)ng * DFEAT;  // B[k][ng] = Wl[ng*D + k]
    const float* wr_row = Wr + (long)ng * DFEAT;

    v8f c = {};
    #pragma unroll
    for (int k0 = 0; k0 < DFEAT; k0 += 4) {
        const int kk = k0 + 2 * half;
        v2f a1 = *reinterpret_cast<const v2f*>(&sA[mn * DFEAT + kk]);
        v2f b1 = *reinterpret_cast<const v2f*>(wl_row + kk);
        c = __builtin_amdgcn_wmma_f32_16x16x4_f32(false, a1, false, b1,
                                                  (short)0, c, false, false);
        v2f a2 = *reinterpret_cast<const v2f*>(&sH[mn * DFEAT + kk]);
        v2f b2 = *reinterpret_cast<const v2f*>(wr_row + kk);
        c = __builtin_amdgcn_wmma_f32_16x16x4_f32(false, a2, false, b2,
                                                  (short)0, c, false, false);
    }

    // Epilogue: bias, store pre-BN output, accumulate BN statistics.
    const float bias = bl[ng];
    float s = 0.0f, q = 0.0f;
    #pragma unroll
    for (int v = 0; v < 8; ++v) {
        float val = c[v] + bias;
        out[(row0 + v + 8 * half) * DFEAT + ng] = val;
        s += val;
        q += val * val;
    }
    atomicAdd(&bnsum[ng], s);
    atomicAdd(&bnsq[ng],  q);
}

// ---------------------------------------------------------------------------
// Kernel 3: BatchNorm finalize (in place on d_out).
// ---------------------------------------------------------------------------
__global__ void __launch_bounds__(256) bn_finalize_kernel(
    float* __restrict__ out, const float* __restrict__ bnsum,
    const float* __restrict__ bnsq, const float* __restrict__ gamma,
    const float* __restrict__ beta)
{
    long i = (long)blockIdx.x * blockDim.x + threadIdx.x;
    if (i >= (long)N_NODES * DFEAT) return;
    int col = (int)(i & (DFEAT - 1));
    const float invN = 1.0f / (float)N_NODES;
    float mu  = bnsum[col] * invN;
    float var = bnsq[col] * invN - mu * mu;
    float sc  = gamma[col] * rsqrtf(var + BN_EPS);
    out[i] = (out[i] - mu) * sc + beta[col];
}

// ---------------------------------------------------------------------------
// Launch
// ---------------------------------------------------------------------------
extern "C" void kernel_launch(void* const* d_in, const int* in_sizes, int n_in,
                              void* d_out, int out_size, void* d_ws, size_t ws_size,
                              hipStream_t stream) {
    const float* x     = (const float*)d_in[0];
    // d_in[1] = edge_attr: unused by the reference
    const float* Wl    = (const float*)d_in[2];
    const float* bl    = (const float*)d_in[3];
    const float* Wr    = (const float*)d_in[4];
    const float* gamma = (const float*)d_in[5];
    const float* beta  = (const float*)d_in[6];
    const int*   ei    = (const int*)d_in[7];
    float* out = (float*)d_out;

    // Workspace layout: agg[N*D] | cnt[N] | bnsum[D] | bnsq[D]
    float* agg   = (float*)d_ws;
    float* cnt   = agg + (size_t)N_NODES * DFEAT;
    float* bnsum = cnt + N_NODES;
    float* bnsq  = bnsum + DFEAT;

    const long zn = (long)N_NODES * DFEAT + N_NODES + 2 * DFEAT;
    zero_ws_kernel<<<(int)((zn + 255) / 256), 256, 0, stream>>>((float*)d_ws, zn);

    const long sthreads = (long)N_EDGES * 32;
    scatter_edges_kernel<<<(int)((sthreads + 255) / 256), 256, 0, stream>>>(x, ei, agg, cnt);

    sage_gemm_kernel<<<N_NODES / 16, 256, 0, stream>>>(x, Wl, bl, Wr, agg, cnt,
                                                       out, bnsum, bnsq);

    const long on = (long)N_NODES * DFEAT;
    bn_finalize_kernel<<<(int)((on + 255) / 256), 256, 0, stream>>>(out, bnsum, bnsq,
                                                                    gamma, beta);
}